// SpGAT_51213190037830
// MI455X (gfx1250) — compile-verified
//
#include <hip/hip_runtime.h>
#include <hip/hip_bf16.h>

typedef float v2f __attribute__((ext_vector_type(2)));
typedef float v8f __attribute__((ext_vector_type(8)));

#define GAT_D 256
#define ALPHA 0.2f

// ---------------------------------------------------------------------------
// WMMA fp32 GEMM: H[b] = A[b] (NxD) @ W (DxD), row-major everywhere.
// grid = (N/16, B), block = 128 (4 waves). Wave w computes rows
// [mtile*16, mtile*16+16) x cols [w*64, w*64+64) as four 16x16 WMMA tiles.
// A-frag layout (32-bit A 16x4): lanes 0-15 hold M=lane, VGPR0=K0/ VGPR1=K1;
// lanes 16-31 hold M=lane-16, K2/K3. B mirrors with N on lanes.
// C/D layout: VGPR j -> (M = j + 8*half, N = lane&15).
// ---------------------------------------------------------------------------
__global__ __launch_bounds__(128) void gat_gemm_f32(
    const float* __restrict__ A, const float* __restrict__ W,
    float* __restrict__ H, int N) {
  const int D = GAT_D;
  const int mtile = blockIdx.x;
  const int b = blockIdx.y;
  const float* Ab = A + (size_t)b * N * D;
  float* Hb = H + (size_t)b * N * D;

  const int wave = threadIdx.x >> 5;
  const int lane = threadIdx.x & 31;
  const int half = lane >> 4;
  const int l16 = lane & 15;
  const int colbase = wave * 64;

  const float* Arow = Ab + (size_t)(mtile * 16 + l16) * D;

  v8f c0 = {}, c1 = {}, c2 = {}, c3 = {};

  for (int k = 0; k < D; k += 4) {
    const int ka = k + half * 2;          // this lane-half's K pair
    v2f a;
    a.x = Arow[ka];
    a.y = Arow[ka + 1];

    const float* Wk = W + (size_t)ka * D;
    const int cc = colbase + l16;
    v2f b0, b1, b2, b3;
    b0.x = Wk[cc +  0]; b0.y = Wk[D + cc +  0];
    b1.x = Wk[cc + 16]; b1.y = Wk[D + cc + 16];
    b2.x = Wk[cc + 32]; b2.y = Wk[D + cc + 32];
    b3.x = Wk[cc + 48]; b3.y = Wk[D + cc + 48];

    c0 = __builtin_amdgcn_wmma_f32_16x16x4_f32(false, a, false, b0, (short)0, c0, false, false);
    c1 = __builtin_amdgcn_wmma_f32_16x16x4_f32(false, a, false, b1, (short)0, c1, false, false);
    c2 = __builtin_amdgcn_wmma_f32_16x16x4_f32(false, a, false, b2, (short)0, c2, false, false);
    c3 = __builtin_amdgcn_wmma_f32_16x16x4_f32(false, a, false, b3, (short)0, c3, false, false);
  }

  const int rbase = mtile * 16 + half * 8;
#pragma unroll
  for (int j = 0; j < 8; ++j) {
    float* Hr = Hb + (size_t)(rbase + j) * D + l16;
    Hr[colbase +  0] = c0[j];
    Hr[colbase + 16] = c1[j];
    Hr[colbase + 32] = c2[j];
    Hr[colbase + 48] = c3[j];
  }
}

// ---------------------------------------------------------------------------
// out[r] = dot(M[r, 0:256], v[0:256]); one wave32 per row, float4 loads,
// shuffle-xor reduce. block = 256 (8 waves).
// ---------------------------------------------------------------------------
__global__ __launch_bounds__(256) void gat_rowdot(
    const float* __restrict__ M, const float* __restrict__ v,
    float* __restrict__ out, int rows) {
  const int row = blockIdx.x * (blockDim.x >> 5) + (threadIdx.x >> 5);
  if (row >= rows) return;
  const int lane = threadIdx.x & 31;
  const float4* r = (const float4*)(M + (size_t)row * GAT_D);
  const float4* vv = (const float4*)v;
  float4 m0 = r[lane * 2 + 0];
  float4 m1 = r[lane * 2 + 1];
  float4 v0 = vv[lane * 2 + 0];
  float4 v1 = vv[lane * 2 + 1];
  float p = m0.x * v0.x + m0.y * v0.y + m0.z * v0.z + m0.w * v0.w +
            m1.x * v1.x + m1.y * v1.y + m1.z * v1.z + m1.w * v1.w;
#pragma unroll
  for (int off = 16; off > 0; off >>= 1) p += __shfl_xor(p, off, 32);
  if (lane == 0) out[row] = p;
}

// ---------------------------------------------------------------------------
// Edge scatter: one wave32 per edge. w = exp(-leakyrelu(sq[src]+sn[dst])).
// Each lane atomically accumulates 8 features of w*h[dst] into agg[src];
// lane 0 accumulates rowsum[src]. grid = (E/8, B), block = 256.
// ---------------------------------------------------------------------------
__global__ __launch_bounds__(256) void gat_edge(
    const int* __restrict__ edges, const float* __restrict__ sq,
    const float* __restrict__ sn, const float* __restrict__ H,
    float* __restrict__ agg, float* __restrict__ rowsum, int N, int E) {
  const int b = blockIdx.y;
  const int e = blockIdx.x * (blockDim.x >> 5) + (threadIdx.x >> 5);
  if (e >= E) return;
  const int lane = threadIdx.x & 31;
  const int* eb = edges + (size_t)b * 2 * E;
  const int src = eb[e];
  const int dst = eb[E + e];

  const float s = sq[(size_t)b * N + src] + sn[(size_t)b * N + dst];
  const float lr = (s >= 0.0f) ? s : ALPHA * s;
  const float w = expf(-lr);

  const float* hd = H + ((size_t)b * N + dst) * GAT_D;
  float* ag = agg + ((size_t)b * N + src) * GAT_D;
  const int d0 = lane * 8;
  float4 h0 = *(const float4*)(hd + d0);
  float4 h1 = *(const float4*)(hd + d0 + 4);
  atomicAdd(ag + d0 + 0, w * h0.x);
  atomicAdd(ag + d0 + 1, w * h0.y);
  atomicAdd(ag + d0 + 2, w * h0.z);
  atomicAdd(ag + d0 + 3, w * h0.w);
  atomicAdd(ag + d0 + 4, w * h1.x);
  atomicAdd(ag + d0 + 5, w * h1.y);
  atomicAdd(ag + d0 + 6, w * h1.z);
  atomicAdd(ag + d0 + 7, w * h1.w);
  if (lane == 0) atomicAdd(rowsum + (size_t)b * N + src, w);
}

// ---------------------------------------------------------------------------
// out = (rowsum==0) ? 0 : agg/rowsum  [+ residual]. float4-vectorized.
// i indexes float4 units; 64 float4 per row of 256 features.
// ---------------------------------------------------------------------------
__global__ __launch_bounds__(256) void gat_norm(
    const float* __restrict__ agg, const float* __restrict__ rowsum,
    const float* __restrict__ residual, float* __restrict__ out, long total4) {
  const long i = (long)blockIdx.x * blockDim.x + threadIdx.x;
  if (i >= total4) return;
  const long row = i >> 6;
  const float r = rowsum[row];
  float4 v;
  if (r == 0.0f) {
    v.x = v.y = v.z = v.w = 0.0f;
  } else {
    const float inv = 1.0f / r;
    float4 a = ((const float4*)agg)[i];
    v.x = a.x * inv; v.y = a.y * inv; v.z = a.z * inv; v.w = a.w * inv;
  }
  if (residual) {
    float4 n = ((const float4*)residual)[i];
    v.x += n.x; v.y += n.y; v.z += n.z; v.w += n.w;
  }
  ((float4*)out)[i] = v;
}

extern "C" void kernel_launch(void* const* d_in, const int* in_sizes, int n_in,
                              void* d_out, int out_size, void* d_ws, size_t ws_size,
                              hipStream_t stream) {
  const float* nodes  = (const float*)d_in[0];   // [B,N,D]
  const float* nodesq = (const float*)d_in[1];   // [B,N,D]
  const float* W      = (const float*)d_in[2];   // [L,D,D]
  const float* a      = (const float*)d_in[3];   // [L,2D]
  const int*   edges  = (const int*)d_in[4];     // [B,2,E]

  const int D = GAT_D;
  const int B = 2;
  const int L = in_sizes[3] / (2 * D);
  const int N = in_sizes[0] / (B * D);
  const int E = in_sizes[4] / (2 * B);
  const size_t BND = (size_t)B * N * D;
  const size_t BN  = (size_t)B * N;

  float* ws     = (float*)d_ws;
  float* h      = ws;            // [B,N,D]
  float* xbuf   = h + BND;       // [B,N,D] inter-layer activations
  float* agg    = xbuf + BND;    // [B,N,D]
  float* sq     = agg + BND;     // [B,N]
  float* sn     = sq + BN;       // [B,N]
  float* rowsum = sn + BN;       // [B,N]
  float* wa     = rowsum + BN;   // [D]
  float* out    = (float*)d_out;

  const int rowsBN = (int)BN;
  for (int l = 0; l < L; ++l) {
    const float* X  = (l == 0) ? nodes : xbuf;
    const float* Wl = W + (size_t)l * D * D;
    const float* al = a + (size_t)l * 2 * D;

    // wa = Wl @ al[:D]  (folds the hq GEMM into a matvec)
    gat_rowdot<<<(D + 7) / 8, 256, 0, stream>>>(Wl, al, wa, D);
    // h = X @ Wl  (WMMA fp32)
    gat_gemm_f32<<<dim3(N / 16, B), 128, 0, stream>>>(X, Wl, h, N);
    // sq = xq @ wa ; sn = h @ al[D:]
    gat_rowdot<<<(rowsBN + 7) / 8, 256, 0, stream>>>(nodesq, wa, sq, rowsBN);
    gat_rowdot<<<(rowsBN + 7) / 8, 256, 0, stream>>>(h, al + D, sn, rowsBN);

    hipMemsetAsync(agg, 0, BND * sizeof(float), stream);
    hipMemsetAsync(rowsum, 0, BN * sizeof(float), stream);
    gat_edge<<<dim3((E + 7) / 8, B), 256, 0, stream>>>(edges, sq, sn, h, agg,
                                                       rowsum, N, E);

    const bool last = (l == L - 1);
    const long total4 = (long)(BND / 4);
    gat_norm<<<(int)((total4 + 255) / 256), 256, 0, stream>>>(
        agg, rowsum, last ? nodes : nullptr, last ? out : xbuf, total4);
  }
}